// GCN_60129542534
// MI455X (gfx1250) — compile-verified
//
#include <hip/hip_runtime.h>
#include <math.h>

typedef __attribute__((ext_vector_type(2))) float v2f;
typedef __attribute__((ext_vector_type(8))) float v8f;

#define GCN_N_NODES 50000
#define GCN_N_EDGES 600000

// ---------------------------------------------------------------------------
// Utility: zero-fill
// ---------------------------------------------------------------------------
__global__ void gcn_fill_zero(float* __restrict__ p, int n) {
    int i = blockIdx.x * blockDim.x + threadIdx.x;
    if (i < n) p[i] = 0.0f;
}

// ---------------------------------------------------------------------------
// Degree counting: deg_out[src[e]] += 1, deg_in[dst[e]] += 1
// ---------------------------------------------------------------------------
__global__ void gcn_count_deg(const int* __restrict__ src, const int* __restrict__ dst,
                              float* __restrict__ deg_out, float* __restrict__ deg_in,
                              int E) {
    int e = blockIdx.x * blockDim.x + threadIdx.x;
    if (e < E) {
        atomicAdd(&deg_out[src[e]], 1.0f);
        atomicAdd(&deg_in[dst[e]], 1.0f);
    }
}

// d[i] = (max(d[i],1))^{-1/2}
__global__ void gcn_inv_sqrt_clamp(float* __restrict__ d, int n) {
    int i = blockIdx.x * blockDim.x + threadIdx.x;
    if (i < n) {
        float v = d[i];
        v = v < 1.0f ? 1.0f : v;
        d[i] = 1.0f / sqrtf(v);
    }
}

// ---------------------------------------------------------------------------
// Edge scatter: agg[dst[e], :] += x[src[e], :] * deg_out_is[src[e]]
// 32 threads per edge, each handling a float4 chunk (d = 128).
// ---------------------------------------------------------------------------
__global__ void gcn_scatter(const float* __restrict__ x,
                            const int* __restrict__ src, const int* __restrict__ dst,
                            const float* __restrict__ deg_out_is,
                            float* __restrict__ agg, int E) {
    const int CH = 32;  // 128 floats / 4 per chunk
    int t = blockIdx.x * blockDim.x + threadIdx.x;
    int e = t >> 5;          // t / 32
    if (e >= E) return;
    int c = t & 31;
    int s  = src[e];
    int dd = dst[e];
    float sc = deg_out_is[s];
    const float4 v = *(const float4*)(x + (size_t)s * 128 + c * 4);
    float* base = agg + (size_t)dd * 128 + c * 4;
    atomicAdd(base + 0, v.x * sc);
    atomicAdd(base + 1, v.y * sc);
    atomicAdd(base + 2, v.z * sc);
    atomicAdd(base + 3, v.w * sc);
    (void)CH;
}

// ---------------------------------------------------------------------------
// GEMM: out[m,n] = relu?( sum_k (A[m,k]*deg_in_is[m]) * W[k,n] + bias[n] )
// One wave32 per 16x16 output tile, K=128, C in {128,16}.
// Uses V_WMMA_F32_16X16X4_F32 (fp32 matrix pipe) — 32 WMMA ops per tile.
// ---------------------------------------------------------------------------
__global__ void gcn_gemm_wmma(const float* __restrict__ A,
                              const float* __restrict__ deg_in_is,
                              const float* __restrict__ W,
                              const float* __restrict__ bias,
                              float* __restrict__ out,
                              int K, int C, int do_relu) {
    const int m0   = blockIdx.x * 16;
    const int n0   = blockIdx.y * 16;
    const int lane = threadIdx.x & 31;
    const int half = lane >> 4;     // 0: K+0/1 ; 1: K+2/3
    const int l15  = lane & 15;
    const int m = m0 + l15;         // A-fragment row for this lane
    const int n = n0 + l15;         // B-fragment col for this lane

    const float scale = deg_in_is[m];
    const float* arow = A + (size_t)m * K;

    v8f c = {};  // zero accumulator
    for (int k0 = 0; k0 < K; k0 += 4) {
        const int kb = k0 + half * 2;
        v2f a, b;
        a.x = arow[kb]     * scale;
        a.y = arow[kb + 1] * scale;
        b.x = W[(size_t)kb       * C + n];
        b.y = W[(size_t)(kb + 1) * C + n];
        // 8 args: (neg_a, A, neg_b, B, c_mod, C, reuse_a, reuse_b)
        c = __builtin_amdgcn_wmma_f32_16x16x4_f32(
                false, a, false, b, (short)0, c, false, false);
    }

    const float bv = bias[n];
#pragma unroll
    for (int v = 0; v < 8; ++v) {
        float val = c[v] + bv;
        if (do_relu) val = fmaxf(val, 0.0f);
        out[(size_t)(m0 + half * 8 + v) * C + n] = val;
    }
}

// ---------------------------------------------------------------------------
// Host-side orchestration
// ---------------------------------------------------------------------------
extern "C" void kernel_launch(void* const* d_in, const int* in_sizes, int n_in,
                              void* d_out, int out_size, void* d_ws, size_t ws_size,
                              hipStream_t stream) {
    const float* features = (const float*)d_in[0];  // [N,128]
    const int*   src      = (const int*)  d_in[1];  // [E]
    const int*   dst      = (const int*)  d_in[2];  // [E]
    const float* W1       = (const float*)d_in[3];  // [128,128]
    const float* b1       = (const float*)d_in[4];  // [128]
    const float* W2       = (const float*)d_in[5];  // [128,128]
    const float* b2       = (const float*)d_in[6];  // [128]
    const float* W3       = (const float*)d_in[7];  // [128,16]
    const float* b3       = (const float*)d_in[8];  // [16]
    float* out = (float*)d_out;                     // [N,16]

    const int N = GCN_N_NODES;
    const int E = GCN_N_EDGES;
    const size_t NF = (size_t)N * 128;

    float* bufA  = (float*)d_ws;          // [N,128] agg
    float* bufB  = bufA + NF;             // [N,128] layer activations
    float* deg_o = bufB + NF;             // [N] deg_out^{-1/2}
    float* deg_i = deg_o + N;             // [N] deg_in^{-1/2}

    const int TB = 256;

    // --- degrees ---
    gcn_fill_zero<<<(2 * N + TB - 1) / TB, TB, 0, stream>>>(deg_o, 2 * N);
    gcn_count_deg<<<(E + TB - 1) / TB, TB, 0, stream>>>(src, dst, deg_o, deg_i, E);
    gcn_inv_sqrt_clamp<<<(2 * N + TB - 1) / TB, TB, 0, stream>>>(deg_o, 2 * N);

    const int scatter_blocks = (E * 32 + TB - 1) / TB;
    const dim3 gemm128(N / 16, 128 / 16);  // 3125 x 8 waves
    const dim3 gemm16 (N / 16, 16 / 16);   // 3125 x 1 waves

    // --- layer 1: features -> bufB (relu) ---
    gcn_fill_zero<<<((int)NF + TB - 1) / TB, TB, 0, stream>>>(bufA, (int)NF);
    gcn_scatter<<<scatter_blocks, TB, 0, stream>>>(features, src, dst, deg_o, bufA, E);
    gcn_gemm_wmma<<<gemm128, 32, 0, stream>>>(bufA, deg_i, W1, b1, bufB, 128, 128, 1);

    // --- layer 2: bufB -> bufB (relu) ---
    gcn_fill_zero<<<((int)NF + TB - 1) / TB, TB, 0, stream>>>(bufA, (int)NF);
    gcn_scatter<<<scatter_blocks, TB, 0, stream>>>(bufB, src, dst, deg_o, bufA, E);
    gcn_gemm_wmma<<<gemm128, 32, 0, stream>>>(bufA, deg_i, W2, b2, bufB, 128, 128, 1);

    // --- layer 3: bufB -> out (no relu) ---
    gcn_fill_zero<<<((int)NF + TB - 1) / TB, TB, 0, stream>>>(bufA, (int)NF);
    gcn_scatter<<<scatter_blocks, TB, 0, stream>>>(bufB, src, dst, deg_o, bufA, E);
    gcn_gemm_wmma<<<gemm16, 32, 0, stream>>>(bufA, deg_i, W3, b3, out, 128, 16, 0);

    (void)in_sizes; (void)n_in; (void)out_size; (void)ws_size;
}